// Brick_Wall_Network_71408126263790
// MI455X (gfx1250) — compile-verified
//
#include <hip/hip_runtime.h>

typedef __attribute__((ext_vector_type(2))) float v2f;
typedef __attribute__((ext_vector_type(8))) float v8f;

#define BCOLS 8192

// ---------------------------------------------------------------------------
// Kernel 1: build Ur = Re(L5 L4 L3 L2 L1)  (16x16 real), L2..L5 are row
// permutations, so Ur[r,c] = Re(layer1[sigma(r), c]).
// sigma = pi2 o pi3 o pi4 o pi5 applied to r (bit3=q0 ... bit0=q3):
//   pi5: CNOT(q1->q3), pi4: CNOT(q2->q3), pi3: CNOT(q1->q2), pi2: CNOT(q0->q1)
// ---------------------------------------------------------------------------
__device__ inline void u_entry(float phi, float theta, float omega,
                               int i, int j, float& re, float& im) {
    float h  = theta * 0.5f;
    float c  = __cosf(h), s = __sinf(h);
    float po = (phi + omega) * 0.5f;
    float pm = (phi - omega) * 0.5f;
    if (i == 0 && j == 0)      { re =  c * __cosf(po); im = -c * __sinf(po); }
    else if (i == 0 && j == 1) { re = -s * __cosf(pm); im = -s * __sinf(pm); }
    else if (i == 1 && j == 0) { re =  s * __cosf(pm); im = -s * __sinf(pm); }
    else                       { re =  c * __cosf(po); im =  c * __sinf(po); }
}

__global__ void build_ur_kernel(
    const float* __restrict__ p1,  const float* __restrict__ t1,  const float* __restrict__ o1,
    const float* __restrict__ p2,  const float* __restrict__ t2,  const float* __restrict__ o2,
    const float* __restrict__ p3,  const float* __restrict__ t3,  const float* __restrict__ o3,
    const float* __restrict__ p4,  const float* __restrict__ t4,  const float* __restrict__ o4,
    float* __restrict__ Ur) {
    int t = threadIdx.x;           // 256 threads: t = r*16 + c
    int r = t >> 4, c = t & 15;
    // row permutation sigma(r)
    int q = r;
    if ((q >> 2) & 1) q ^= 1;      // pi5: CNOT q1->q3
    if ((q >> 1) & 1) q ^= 1;      // pi4: CNOT q2->q3
    if ((q >> 2) & 1) q ^= 2;      // pi3: CNOT q1->q2
    if ((q >> 3) & 1) q ^= 4;      // pi2: CNOT q0->q1
    // gate rows/cols from bits of q / c (qubit0 = msb)
    int r0 = (q >> 3) & 1, r1 = (q >> 2) & 1, r2 = (q >> 1) & 1, r3 = q & 1;
    int c0 = (c >> 3) & 1, c1 = (c >> 2) & 1, c2 = (c >> 1) & 1, c3 = c & 1;
    float re, im, gr, gi;
    // g1(phi1,theta1,omega1)
    u_entry(p1[0], t1[0], o1[0], r0, c0, re, im);
    // g2(phi2, theta3, omega4)  -- intentional param mix in the reference
    u_entry(p2[0], t3[0], o4[0], r1, c1, gr, gi);
    { float nr = re * gr - im * gi, ni = re * gi + im * gr; re = nr; im = ni; }
    // g3(phi3, theta3, omega3)
    u_entry(p3[0], t3[0], o3[0], r2, c2, gr, gi);
    { float nr = re * gr - im * gi, ni = re * gi + im * gr; re = nr; im = ni; }
    // g4(phi4, theta4, omega4)
    u_entry(p4[0], t4[0], o4[0], r3, c3, gr, gi);
    { float nr = re * gr - im * gi; re = nr; (void)t2; }
    Ur[r * 16 + c] = re;
    (void)im;
}

// ---------------------------------------------------------------------------
// Kernel 2: a = Ur @ x   (16 x 8192), fully coalesced over columns.
// ---------------------------------------------------------------------------
__global__ __launch_bounds__(256) void project_kernel(
    const float* __restrict__ x, const float* __restrict__ Ur,
    float* __restrict__ aBuf) {
    __shared__ float sU[256];
    int t = threadIdx.x;
    sU[t] = Ur[t];
    __syncthreads();
    int col = blockIdx.x * 256 + t;
    float xv[16];
#pragma unroll
    for (int k = 0; k < 16; ++k) xv[k] = x[k * BCOLS + col];
#pragma unroll
    for (int r = 0; r < 16; ++r) {
        float acc = 0.f;
#pragma unroll
        for (int k = 0; k < 16; ++k) acc = fmaf(sU[r * 16 + k], xv[k], acc);
        aBuf[r * BCOLS + col] = acc;
    }
}

// ---------------------------------------------------------------------------
// Kernel 3: out = a^T (M a), rank-16 Gram matrix via V_WMMA_F32_16X16X4_F32.
// One wave -> 16x64 output strip (4 tiles share the A fragments).
// M = diag(+1,-1,...): folded into B fragment as component-parity sign
// (k0 = kc*4 + 2*half is always even => .x is +, .y is -).
// Fragment layouts per ISA 7.12.2:
//   A 16x4 f32 : lane<16: M=lane, VGPRv=K v ; lane>=16: M=lane-16, VGPRv=K 2+v
//   B  4x16 f32: lane<16: N=lane, VGPRv=K v ; lane>=16: N=lane-16, VGPRv=K 2+v
//   D 16x16 f32: VGPRv: row v + 8*(lane>>4), col = lane&15
// ---------------------------------------------------------------------------
__global__ __launch_bounds__(256) void gram_kernel(
    const float* __restrict__ aBuf, float* __restrict__ out) {
    const int wave = threadIdx.x >> 5;
    const int lane = threadIdx.x & 31;
    const int half = lane >> 4;
    const int mn   = lane & 15;

    const unsigned g   = blockIdx.x * 8u + wave;  // 512 * 128 = 65536 strips
    const int i0 = (int)(g >> 7) * 16;            // 512 i-tiles
    const int j0 = (int)(g & 127u) * 64;          // 128 j-strips of 64

    v8f acc0 = {}, acc1 = {}, acc2 = {}, acc3 = {};
#pragma unroll
    for (int kc = 0; kc < 4; ++kc) {
        const int k0 = kc * 4 + half * 2;         // even
        const float* rowA0 = aBuf + (k0 + 0) * BCOLS;
        const float* rowA1 = aBuf + (k0 + 1) * BCOLS;
        v2f af;
        af.x = rowA0[i0 + mn];
        af.y = rowA1[i0 + mn];
        v2f b0, b1, b2, b3;
        b0.x =  rowA0[j0 +  0 + mn];  b0.y = -rowA1[j0 +  0 + mn];
        b1.x =  rowA0[j0 + 16 + mn];  b1.y = -rowA1[j0 + 16 + mn];
        b2.x =  rowA0[j0 + 32 + mn];  b2.y = -rowA1[j0 + 32 + mn];
        b3.x =  rowA0[j0 + 48 + mn];  b3.y = -rowA1[j0 + 48 + mn];
        acc0 = __builtin_amdgcn_wmma_f32_16x16x4_f32(false, af, false, b0, (short)0, acc0, false, false);
        acc1 = __builtin_amdgcn_wmma_f32_16x16x4_f32(false, af, false, b1, (short)0, acc1, false, false);
        acc2 = __builtin_amdgcn_wmma_f32_16x16x4_f32(false, af, false, b2, (short)0, acc2, false, false);
        acc3 = __builtin_amdgcn_wmma_f32_16x16x4_f32(false, af, false, b3, (short)0, acc3, false, false);
    }
#pragma unroll
    for (int v = 0; v < 8; ++v) {
        size_t row = (size_t)(i0 + v + 8 * half) * BCOLS;
        __builtin_nontemporal_store(acc0[v], &out[row + j0 +  0 + mn]);
        __builtin_nontemporal_store(acc1[v], &out[row + j0 + 16 + mn]);
        __builtin_nontemporal_store(acc2[v], &out[row + j0 + 32 + mn]);
        __builtin_nontemporal_store(acc3[v], &out[row + j0 + 48 + mn]);
    }
}

// ---------------------------------------------------------------------------
extern "C" void kernel_launch(void* const* d_in, const int* in_sizes, int n_in,
                              void* d_out, int out_size, void* d_ws, size_t ws_size,
                              hipStream_t stream) {
    const float* x = (const float*)d_in[0];
    const float* p[12];
    for (int i = 0; i < 12; ++i) p[i] = (const float*)d_in[i + 1];
    // p[0]=phi1 p[1]=th1 p[2]=om1 p[3]=phi2 p[4]=th2 p[5]=om2
    // p[6]=phi3 p[7]=th3 p[8]=om3 p[9]=phi4 p[10]=th4 p[11]=om4

    float* Ur   = (float*)d_ws;          // 256 floats
    float* aBuf = Ur + 256;              // 16*8192 floats (512 KB)
    float* out  = (float*)d_out;         // 8192*8192 floats

    build_ur_kernel<<<1, 256, 0, stream>>>(
        p[0], p[1], p[2], p[3], p[4], p[5], p[6], p[7], p[8], p[9], p[10], p[11], Ur);
    project_kernel<<<BCOLS / 256, 256, 0, stream>>>(x, Ur, aBuf);
    gram_kernel<<<(512 * 128) / 8, 256, 0, stream>>>(aBuf, out);

    (void)in_sizes; (void)n_in; (void)out_size; (void)ws_size;
}